// KAAsparse_29094108463106
// MI455X (gfx1250) — compile-verified
//
#include <hip/hip_runtime.h>
#include <math.h>

typedef __attribute__((ext_vector_type(2))) float v2f;
typedef __attribute__((ext_vector_type(8))) float v8f;

#define KA 16
#define UF 4   // chunks (of 4 K-elements) per unrolled main-loop iteration

// ---------- monotone float<->uint encoding for deterministic atomicMax ----------
__device__ __forceinline__ unsigned enc_f32(float f) {
    unsigned u = __float_as_uint(f);
    return (u & 0x80000000u) ? ~u : (u | 0x80000000u);
}
__device__ __forceinline__ float dec_f32(unsigned u) {
    return __uint_as_float((u & 0x80000000u) ? (u & 0x7FFFFFFFu) : ~u);
}

// ---------- init: colmax = encoded -inf (0 is the minimum key) ----------
__global__ void init_kernel(unsigned* __restrict__ colmax) {
    if (threadIdx.x < KA) colmax[threadIdx.x] = 0u;
}

// ---------- SsmT[n,a] = softmax over the 16 archetypes of S[:,n] ----------
__global__ void row_softmax_kernel(const float* __restrict__ S,
                                   float* __restrict__ SsmT, int N) {
    int n = blockIdx.x * blockDim.x + threadIdx.x;
    if (n >= N) return;
    float x[KA];
    float mx = -INFINITY;
    for (int a = 0; a < KA; ++a) {
        x[a] = S[a * N + n];                 // coalesced across n for each a
        mx = fmaxf(mx, x[a]);
    }
    float s = 0.f;
    for (int a = 0; a < KA; ++a) { x[a] = expf(x[a] - mx); s += x[a]; }
    float inv = 1.f / s;
    for (int a = 0; a < KA; ++a) SsmT[n * KA + a] = x[a] * inv;
}

// ---------- column max of C (softmax over N, per archetype column) ----------
__global__ void col_max_kernel(const float* __restrict__ C,
                               unsigned* __restrict__ colmax, int N) {
    int total = N * KA;
    int stride = gridDim.x * blockDim.x;            // multiple of 16
    float lm = -INFINITY;
    for (int i = blockIdx.x * blockDim.x + threadIdx.x; i < total; i += stride)
        lm = fmaxf(lm, C[i]);                       // column of i is i&15 == tid&15
    __shared__ float sm[256];
    sm[threadIdx.x] = lm;
    __syncthreads();
    if (threadIdx.x < KA) {
        float m = -INFINITY;
        for (int j = threadIdx.x; j < 256; j += KA) m = fmaxf(m, sm[j]);
        atomicMax(&colmax[threadIdx.x], enc_f32(m)); // order-independent, deterministic
    }
}

// ---------- per-block partial column sums of exp(C - max) ----------
__global__ void col_sum_kernel(const float* __restrict__ C,
                               const unsigned* __restrict__ colmax,
                               float* __restrict__ part, int N) {
    int total = N * KA;
    int stride = gridDim.x * blockDim.x;
    float mx = dec_f32(colmax[threadIdx.x & 15]);
    float ls = 0.f;
    for (int i = blockIdx.x * blockDim.x + threadIdx.x; i < total; i += stride)
        ls += expf(C[i] - mx);
    __shared__ float sm[256];
    sm[threadIdx.x] = ls;
    __syncthreads();
    if (threadIdx.x < KA) {
        float s = 0.f;
        for (int j = threadIdx.x; j < 256; j += KA) s += sm[j];
        part[blockIdx.x * KA + threadIdx.x] = s;
    }
}

// ---------- deterministic fixed-order reduce of column-sum partials ----------
__global__ void col_reduce_kernel(const float* __restrict__ part,
                                  float* __restrict__ colsum, int RB) {
    int t = threadIdx.x;
    if (t >= KA) return;
    float s = 0.f;
    for (int b = 0; b < RB; ++b) s += part[b * KA + t];
    colsum[t] = s;
}

// ---------- Csm[n,a] = exp(C[n,a]-max_a)/sum_a ----------
__global__ void csm_kernel(const float* __restrict__ C,
                           const unsigned* __restrict__ colmax,
                           const float* __restrict__ colsum,
                           float* __restrict__ Csm, int N) {
    int total = N * KA;
    int stride = gridDim.x * blockDim.x;
    int col = threadIdx.x & 15;                     // stride multiple of 16
    float mx = dec_f32(colmax[col]);
    float inv = 1.f / colsum[col];
    for (int i = blockIdx.x * blockDim.x + threadIdx.x; i < total; i += stride)
        Csm[i] = expf(C[i] - mx) * inv;
}

// ---------- G = SsmT^T @ SsmT via V_WMMA_F32_16X16X4_F32 over node chunks of 4 ----------
__global__ void g_wmma_kernel(const float* __restrict__ SsmT,
                              float* __restrict__ Gpart, int N) {
    const int lane = threadIdx.x & 31;
    const int wv   = threadIdx.x >> 5;              // 8 waves / block
    const int half = lane >> 4;
    const int m    = lane & 15;

    int nWaves     = (gridDim.x * blockDim.x) >> 5;
    int gw         = (blockIdx.x * blockDim.x + threadIdx.x) >> 5;
    int nChunksAll = (N + 3) >> 2;
    int nGroups    = (N >> 2) / UF;                 // full, guard-free groups

    v8f acc0 = {};
    v8f acc1 = {};

    // main loop: no per-lane guards, EXEC stays all-ones
    for (int g = gw; g < nGroups; g += nWaves) {
        v2f A[UF];
#pragma unroll
        for (int u = 0; u < UF; ++u) {
            int n0 = (g * UF + u) * 4 + 2 * half;
            A[u].x = SsmT[n0 * KA + m];             // coalesced 64B row reads
            A[u].y = SsmT[(n0 + 1) * KA + m];
        }
#pragma unroll
        for (int u = 0; u < UF; ++u) {
            if (u & 1)
                acc1 = __builtin_amdgcn_wmma_f32_16x16x4_f32(
                    false, A[u], false, A[u], (short)0, acc1, false, false);
            else
                acc0 = __builtin_amdgcn_wmma_f32_16x16x4_f32(
                    false, A[u], false, A[u], (short)0, acc0, false, false);
        }
    }

    // tail: leftover chunks incl. partial final chunk (guarded loads)
    for (int ch = nGroups * UF + gw; ch < nChunksAll; ch += nWaves) {
        int n0 = ch * 4 + 2 * half;
        int n1 = n0 + 1;
        int i0 = (n0 < N) ? n0 : 0;
        int i1 = (n1 < N) ? n1 : 0;
        v2f A;
        A.x = (n0 < N) ? SsmT[i0 * KA + m] : 0.f;
        A.y = (n1 < N) ? SsmT[i1 * KA + m] : 0.f;
        acc0 = __builtin_amdgcn_wmma_f32_16x16x4_f32(
            false, A, false, A, (short)0, acc0, false, false);
    }

    __shared__ float sm[8 * 256];
    for (int i = 0; i < 8; ++i)
        sm[wv * 256 + (i + 8 * half) * 16 + m] = acc0[i] + acc1[i];  // ISA C/D layout
    __syncthreads();
    float s = 0.f;
    int t = threadIdx.x;
    for (int w = 0; w < 8; ++w) s += sm[w * 256 + t];
    Gpart[blockIdx.x * 256 + t] = s;
}

// ---------- M and Mp accumulated over edges, 4 edges per WMMA ----------
__global__ void edge_wmma_kernel(const int*   __restrict__ rows,
                                 const int*   __restrict__ cols,
                                 const float* __restrict__ vals,
                                 const float* __restrict__ Csm,
                                 const float* __restrict__ SsmT,
                                 float* __restrict__ Mpart,
                                 float* __restrict__ Mppart, int NNZ) {
    const int lane = threadIdx.x & 31;
    const int wv   = threadIdx.x >> 5;
    const int half = lane >> 4;
    const int m    = lane & 15;

    int nWaves     = (gridDim.x * blockDim.x) >> 5;
    int gw         = (blockIdx.x * blockDim.x + threadIdx.x) >> 5;
    int nChunksAll = (NNZ + 3) >> 2;
    int nGroups    = (NNZ >> 2) / UF;               // full, guard-free groups

    const int2*   __restrict__ rows2 = (const int2*)rows;    // pairs (e, e+1)
    const int2*   __restrict__ cols2 = (const int2*)cols;
    const float2* __restrict__ vals2 = (const float2*)vals;

    v8f accM0 = {}, accM1 = {};
    v8f accP0 = {}, accP1 = {};

    // main loop: no per-lane guards, EXEC stays all-ones
    for (int g = gw; g < nGroups; g += nWaves) {
        int2   r[UF], c[UF];
        float2 w[UF];
        int basep = g * UF * 2 + half;              // pair index base
#pragma unroll
        for (int u = 0; u < UF; ++u) {              // batch metadata loads (b64)
            r[u] = rows2[basep + u * 2];
            c[u] = cols2[basep + u * 2];
            w[u] = vals2[basep + u * 2];
        }
#pragma unroll
        for (int u = 0; u < UF; ++u) {
            v2f A, B, Bp;
            A.x  = w[u].x * Csm[r[u].x * KA + m];   // coalesced 64B row gathers (L2)
            A.y  = w[u].y * Csm[r[u].y * KA + m];
            B.x  = Csm[c[u].x * KA + m];
            B.y  = Csm[c[u].y * KA + m];
            Bp.x = SsmT[c[u].x * KA + m];
            Bp.y = SsmT[c[u].y * KA + m];
            if (u & 1) {
                accM1 = __builtin_amdgcn_wmma_f32_16x16x4_f32(
                    false, A, false, B, (short)0, accM1, false, false);
                accP1 = __builtin_amdgcn_wmma_f32_16x16x4_f32(
                    false, A, false, Bp, (short)0, accP1, false, false);
            } else {
                accM0 = __builtin_amdgcn_wmma_f32_16x16x4_f32(
                    false, A, false, B, (short)0, accM0, false, false);
                accP0 = __builtin_amdgcn_wmma_f32_16x16x4_f32(
                    false, A, false, Bp, (short)0, accP0, false, false);
            }
        }
    }

    // tail: leftover chunks incl. partial final chunk (guarded scalar loads)
    for (int ch = nGroups * UF + gw; ch < nChunksAll; ch += nWaves) {
        int e0 = ch * 4 + 2 * half;
        int e1 = e0 + 1;
        int ee0 = (e0 < NNZ) ? e0 : 0;
        int ee1 = (e1 < NNZ) ? e1 : 0;
        float w0 = (e0 < NNZ) ? vals[ee0] : 0.f;    // w=0 zeroes the A column
        float w1 = (e1 < NNZ) ? vals[ee1] : 0.f;
        int r0 = rows[ee0], r1 = rows[ee1];
        int c0 = cols[ee0], c1 = cols[ee1];

        v2f A, B, Bp;
        A.x  = w0 * Csm[r0 * KA + m];
        A.y  = w1 * Csm[r1 * KA + m];
        B.x  = Csm[c0 * KA + m];
        B.y  = Csm[c1 * KA + m];
        Bp.x = SsmT[c0 * KA + m];
        Bp.y = SsmT[c1 * KA + m];

        accM0 = __builtin_amdgcn_wmma_f32_16x16x4_f32(
            false, A, false, B, (short)0, accM0, false, false);
        accP0 = __builtin_amdgcn_wmma_f32_16x16x4_f32(
            false, A, false, Bp, (short)0, accP0, false, false);
    }

    __shared__ float smM[8 * 256];
    __shared__ float smP[8 * 256];
    for (int i = 0; i < 8; ++i) {
        int idx = (i + 8 * half) * 16 + m;
        smM[wv * 256 + idx] = accM0[i] + accM1[i];
        smP[wv * 256 + idx] = accP0[i] + accP1[i];
    }
    __syncthreads();
    float sM = 0.f, sP = 0.f;
    int t = threadIdx.x;
    for (int w = 0; w < 8; ++w) { sM += smM[w * 256 + t]; sP += smP[w * 256 + t]; }
    Mpart[blockIdx.x * 256 + t]  = sM;
    Mppart[blockIdx.x * 256 + t] = sP;
}

// ---------- finalize: result = -2 trace(Mp) + sum_{a,b} M[a,b] G[b,a] ----------
__global__ void finalize_kernel(const float* __restrict__ Gpart,
                                const float* __restrict__ Mpart,
                                const float* __restrict__ Mppart,
                                float* __restrict__ out, int GB, int EB) {
    int t = threadIdx.x;   // 256 threads, t = a*16 + b
    float g = 0.f;
    for (int bblk = 0; bblk < GB; ++bblk) g += Gpart[bblk * 256 + t];
    float mm = 0.f, mp = 0.f;
    for (int bblk = 0; bblk < EB; ++bblk) {
        mm += Mpart[bblk * 256 + t];
        mp += Mppart[bblk * 256 + t];
    }
    __shared__ float Gs[256];
    __shared__ float red[256];
    Gs[t] = g;
    __syncthreads();
    int a = t >> 4, b = t & 15;
    float contrib = mm * Gs[b * 16 + a];            // M[a,b] * G[b,a]
    if (a == b) contrib -= 2.f * mp;                // -2 * trace(Mp)
    red[t] = contrib;
    __syncthreads();
    for (int s = 128; s > 0; s >>= 1) {             // fixed-order tree reduce
        if (t < s) red[t] += red[t + s];
        __syncthreads();
    }
    if (t == 0) out[0] = red[0];
}

extern "C" void kernel_launch(void* const* d_in, const int* in_sizes, int n_in,
                              void* d_out, int out_size, void* d_ws, size_t ws_size,
                              hipStream_t stream) {
    const float* S      = (const float*)d_in[0];   // [16, N]
    const float* C      = (const float*)d_in[1];   // [N, 16]
    const float* K_vals = (const float*)d_in[2];   // [NNZ]
    const int*   K_rows = (const int*)  d_in[3];
    const int*   K_cols = (const int*)  d_in[4];

    const int N   = in_sizes[0] / KA;
    const int NNZ = in_sizes[2];

    const int RB = 512;    // column-sum partial blocks
    const int GB = 256;    // G WMMA blocks
    const int EB = 1024;   // edge WMMA blocks

    // workspace carve-up (256B aligned)
    char* ws = (char*)d_ws;
    size_t off = 0;
    auto carve = [&](size_t bytes) -> char* {
        char* p = ws + off;
        off = (off + bytes + 255) & ~(size_t)255;
        return p;
    };
    float*    SsmT    = (float*)   carve((size_t)N * KA * sizeof(float));
    float*    Csm     = (float*)   carve((size_t)N * KA * sizeof(float));
    unsigned* colmax  = (unsigned*)carve(KA * sizeof(unsigned));
    float*    sumpart = (float*)   carve((size_t)RB * KA * sizeof(float));
    float*    colsum  = (float*)   carve(KA * sizeof(float));
    float*    Gpart   = (float*)   carve((size_t)GB * 256 * sizeof(float));
    float*    Mpart   = (float*)   carve((size_t)EB * 256 * sizeof(float));
    float*    Mppart  = (float*)   carve((size_t)EB * 256 * sizeof(float));
    (void)ws_size;

    init_kernel<<<1, 32, 0, stream>>>(colmax);
    row_softmax_kernel<<<(N + 255) / 256, 256, 0, stream>>>(S, SsmT, N);
    col_max_kernel<<<512, 256, 0, stream>>>(C, colmax, N);
    col_sum_kernel<<<RB, 256, 0, stream>>>(C, colmax, sumpart, N);
    col_reduce_kernel<<<1, 32, 0, stream>>>(sumpart, colsum, RB);
    csm_kernel<<<2048, 256, 0, stream>>>(C, colmax, colsum, Csm, N);
    g_wmma_kernel<<<GB, 256, 0, stream>>>(SsmT, Gpart, N);
    edge_wmma_kernel<<<EB, 256, 0, stream>>>(K_rows, K_cols, K_vals, Csm, SsmT,
                                             Mpart, Mppart, NNZ);
    finalize_kernel<<<1, 256, 0, stream>>>(Gpart, Mpart, Mppart,
                                           (float*)d_out, GB, EB);
}